// CondMul_1340029796953
// MI455X (gfx1250) — compile-verified
//
#include <hip/hip_runtime.h>
#include <hip/hip_bf16.h>

typedef __attribute__((ext_vector_type(2))) float v2f;
typedef __attribute__((ext_vector_type(8))) float v8f;

#define IN_F  32
#define OUT_F 32

// ---------------- pass 1: zero counts ----------------
__global__ void zero_counts_kernel(int* counts, int classes) {
    int i = blockIdx.x * blockDim.x + threadIdx.x;
    if (i < classes) counts[i] = 0;
}

// ---------------- pass 2: histogram ----------------
__global__ void hist_kernel(const int* __restrict__ inds, int* __restrict__ counts, int n) {
    int i = blockIdx.x * blockDim.x + threadIdx.x;
    if (i < n) atomicAdd(&counts[inds[i]], 1);
}

// ---------------- pass 3: exclusive scan over counts (single block) ----------------
__global__ void scan_kernel(const int* __restrict__ counts, int* __restrict__ offsets,
                            int* __restrict__ cursors, int classes) {
    __shared__ int s[1024];
    int t = threadIdx.x;
    int c = (t < classes) ? counts[t] : 0;
    s[t] = c;
    __syncthreads();
    for (int off = 1; off < 1024; off <<= 1) {
        int v = (t >= off) ? s[t - off] : 0;
        __syncthreads();
        s[t] += v;
        __syncthreads();
    }
    if (t < classes) {
        int excl = s[t] - c;
        offsets[t] = excl;
        cursors[t] = excl;
        if (t == classes - 1) offsets[classes] = s[t];
    }
}

// ---------------- pass 4: scatter row ids into expert-grouped order ----------------
__global__ void scatter_kernel(const int* __restrict__ inds, int* __restrict__ cursors,
                               int* __restrict__ rowIdx, int n) {
    int i = blockIdx.x * blockDim.x + threadIdx.x;
    if (i < n) {
        int e = inds[i];
        int p = atomicAdd(&cursors[e], 1);
        rowIdx[p] = i;
    }
}

// ---------------- pass 5: per-expert WMMA compute ----------------
// One workgroup (128 threads = 4 waves) per expert. Weights staged in LDS once,
// B fragments hoisted to registers, 16 rows per wave-tile via V_WMMA_F32_16X16X4_F32.
__launch_bounds__(128)
__global__ void condmul_wmma_kernel(const float* __restrict__ input,
                                    const float* __restrict__ w,
                                    const float* __restrict__ b,
                                    const int* __restrict__ offsets,
                                    const int* __restrict__ rowIdx,
                                    float* __restrict__ out) {
    const int e     = blockIdx.x;
    const int start = offsets[e];
    const int count = offsets[e + 1] - start;
    if (count == 0) return;                    // block-uniform

    __shared__ float wS[IN_F * 33];            // padded pitch: avoid LDS bank conflicts
    __shared__ float bS[OUT_F];

    const int tid = threadIdx.x;
    const float* we = w + (size_t)e * (IN_F * OUT_F);
    for (int i = tid; i < IN_F * OUT_F; i += 128) {
        int row = i >> 5, col = i & 31;
        wS[row * 33 + col] = we[i];
    }
    if (tid < OUT_F) bS[tid] = b[(size_t)e * OUT_F + tid];
    __syncthreads();

    const int lane = tid & 31;                 // wave32
    const int wave = tid >> 5;
    const int half = lane >> 4;                // 0: lanes 0-15, 1: lanes 16-31
    const int nCol = lane & 15;                // N column within half

    // Preload B fragments (K-major, mirrored A layout): b[kb][v] = w[4*kb + 2*half + v][col]
    v2f bf0[8], bf1[8];
#pragma unroll
    for (int kb = 0; kb < 8; ++kb) {
        int k0 = 4 * kb + 2 * half;
        bf0[kb] = v2f{ wS[k0 * 33 + nCol],      wS[(k0 + 1) * 33 + nCol] };
        bf1[kb] = v2f{ wS[k0 * 33 + nCol + 16], wS[(k0 + 1) * 33 + nCol + 16] };
    }
    const float bias0 = bS[nCol];
    const float bias1 = bS[nCol + 16];

    const int numTiles = (count + 15) >> 4;
    for (int t = wave; t < numTiles; t += 4) { // wave-uniform trip count -> EXEC full at WMMA
        const int base = t * 16;

        // A fragment gather: lane's M row = base + nCol (clamped for partial tiles)
        int pa = base + nCol;
        if (pa > count - 1) pa = count - 1;
        const float* arow = input + (size_t)rowIdx[start + pa] * IN_F;

        v8f acc0, acc1;
#pragma unroll
        for (int r = 0; r < 8; ++r) { acc0[r] = bias0; acc1[r] = bias1; }

#pragma unroll
        for (int kb = 0; kb < 8; ++kb) {
            v2f a = *(const v2f*)(arow + 4 * kb + 2 * half);   // 8-byte aligned b64 load
            acc0 = __builtin_amdgcn_wmma_f32_16x16x4_f32(
                false, a, false, bf0[kb], (short)0, acc0, false, false);
            acc1 = __builtin_amdgcn_wmma_f32_16x16x4_f32(
                false, a, false, bf1[kb], (short)0, acc1, false, false);
        }

        // D layout: VGPR r holds M = r + 8*half, N = nCol
#pragma unroll
        for (int r = 0; r < 8; ++r) {
            int p = base + r + 8 * half;
            if (p < count) {                   // per-lane predicate, after all WMMAs
                size_t row = (size_t)rowIdx[start + p];
                out[row * OUT_F + nCol]      = acc0[r];
                out[row * OUT_F + nCol + 16] = acc1[r];
            }
        }
    }
}

extern "C" void kernel_launch(void* const* d_in, const int* in_sizes, int n_in,
                              void* d_out, int out_size, void* d_ws, size_t ws_size,
                              hipStream_t stream) {
    const float* input = (const float*)d_in[0];
    const int*   inds  = (const int*)d_in[1];
    const float* w     = (const float*)d_in[2];
    const float* b     = (const float*)d_in[3];
    float*       out   = (float*)d_out;

    const int n       = in_sizes[1];                 // N rows
    const int classes = in_sizes[3] / OUT_F;         // b is [classes, 1, 32]

    // workspace layout (ints)
    int* counts  = (int*)d_ws;                       // [classes]
    int* offsets = counts + classes;                 // [classes + 1]
    int* cursors = offsets + classes + 1;            // [classes]
    int* rowIdx  = cursors + classes;                // [n]

    zero_counts_kernel<<<(classes + 255) / 256, 256, 0, stream>>>(counts, classes);
    hist_kernel<<<(n + 255) / 256, 256, 0, stream>>>(inds, counts, n);
    scan_kernel<<<1, 1024, 0, stream>>>(counts, offsets, cursors, classes);
    scatter_kernel<<<(n + 255) / 256, 256, 0, stream>>>(inds, cursors, rowIdx, n);
    condmul_wmma_kernel<<<classes, 128, 0, stream>>>(input, w, b, offsets, rowIdx, out);
}